// MultiAffineRegularizationLoss_25984552141319
// MI455X (gfx1250) — compile-verified
//
#include <hip/hip_runtime.h>

typedef __attribute__((ext_vector_type(2))) float v2f;
typedef __attribute__((ext_vector_type(8))) float v8f;

// ---------------------------------------------------------------------------
// One wave32 processes a tile of 16 output pixels (fixed b, channel, row y,
// columns x0..x0+15).  It builds the 16 x Q patch matrix Pt in LDS (Q = kz^2,
// zero-padded), computes R = Pt * KTK with V_WMMA_F32_16X16X4_F32 (K tiled by
// 4, N tiled by 16), then per-pixel loss = sum_q Pt[m][q] * R[m][q].
// KTK = I - A(AtA)^-1 At is built analytically in LDS by every block.
// Block partials are written to d_ws; a final single-block kernel reduces them
// deterministically into d_out[0].
// ---------------------------------------------------------------------------

template <int KZ>
__global__ __launch_bounds__(256) void affine_loss_kernel(
    const float* __restrict__ flow, int B, int H, int W,
    float* __restrict__ partials, float inv_count)
{
    constexpr int Q  = KZ * KZ;              // patch length (9 or 25)
    constexpr int Qp = ((Q + 3) / 4) * 4;    // K padded to mult of 4 (12 / 28)
    constexpr int NT = (Q + 15) / 16;        // N tiles (1 / 2)
    constexpr int Np = NT * 16;              // N padded (16 / 32)
    constexpr int PS = Np + 2;               // LDS row stride for patch tiles
    constexpr int CK = Qp / 4;               // K chunks (3 / 7)

    const int h = H - KZ + 1;
    const int w = W - KZ + 1;
    const int tilesX = (w + 15) >> 4;
    const long wavesTotal = (long)B * 2 * h * tilesX;

    __shared__ float sKTK[Qp * Np];          // KTK, zero padded
    __shared__ float sP[8 * 16 * PS];        // per-wave 16 x PS patch tiles
    __shared__ float sRed[256];

    const int tid       = threadIdx.x;
    const int lane      = tid & 31;
    const int waveInBlk = tid >> 5;

    // ---- Phase 1: build KTK = I - P,  P_qr = u_q^T (AtA)^-1 u_r ----------
    {
        // AtA = [[p,q,r],[q,p,r],[r,r,n]] with closed-form sums over the grid
        const double p  = (double)KZ * ((double)(KZ - 1) * KZ * (2 * KZ - 1)) / 6.0;
        const double r  = (double)KZ * ((double)KZ * (KZ - 1)) / 2.0;
        const double qv = ((double)KZ * (KZ - 1) / 2.0) * ((double)KZ * (KZ - 1) / 2.0);
        const double nn = (double)Q;
        // symmetric 3x3 inverse via cofactors
        const double m00 = p * nn - r * r;
        const double m01 = r * r - qv * nn;
        const double m02 = qv * r - p * r;
        const double m11 = p * nn - r * r;
        const double m12 = qv * r - p * r;
        const double m22 = p * p - qv * qv;
        const double det = p * m00 + qv * m01 + r * m02;
        const double i00 = m00 / det, i01 = m01 / det, i02 = m02 / det;
        const double i11 = m11 / det, i12 = m12 / det, i22 = m22 / det;

        for (int idx = tid; idx < Qp * Np; idx += 256) {
            const int kq = idx / Np;   // row (q)
            const int nr = idx % Np;   // col (r)
            float val = 0.f;
            if (kq < Q && nr < Q) {
                const double aq = (double)(kq / KZ), cq = (double)(kq % KZ);
                const double ar = (double)(nr / KZ), cr = (double)(nr % KZ);
                const double t0 = i00 * ar + i01 * cr + i02;
                const double t1 = i01 * ar + i11 * cr + i12;
                const double t2 = i02 * ar + i12 * cr + i22;
                const double Pqr = aq * t0 + cq * t1 + t2;
                val = (float)(((kq == nr) ? 1.0 : 0.0) - Pqr);
            }
            sKTK[idx] = val;
        }
    }
    __syncthreads();

    // ---- Phase 2: stage this wave's 16-pixel patch tile into LDS ---------
    const long waveId = (long)blockIdx.x * 8 + waveInBlk;
    const bool waveValid = (waveId < wavesTotal);
    int b = 0, ch = 0, y = 0, x0 = 0;
    {
        long t = waveValid ? waveId : 0;
        x0 = (int)(t % tilesX) * 16;  t /= tilesX;
        y  = (int)(t % h);            t /= h;
        ch = (int)(t % 2);
        b  = (int)(t / 2);
    }
    float* Pw = &sP[waveInBlk * 16 * PS];
    for (int idx = lane; idx < 16 * PS; idx += 32) {
        const int m = idx / PS;        // pixel within tile
        const int k = idx % PS;        // patch component q (padded)
        const int xp = x0 + m;
        float val = 0.f;
        if (waveValid && k < Q && xp < w) {
            const int a = k / KZ, c = k % KZ;
            const int row = y + a, col = xp + c;
            const size_t off = (((size_t)b * 2 + ch) * (size_t)H + row) * (size_t)W + col;
            val = flow[off] + (ch == 0 ? (float)col : (float)row);
        }
        Pw[idx] = val;
    }
    __syncthreads();

    // ---- Phase 3: R = Pt * KTK via WMMA, then loss = <Pt, R> -------------
    float partial = 0.f;
    if (waveValid) {                           // uniform per wave: EXEC all-1s
        const int mA   = lane & 15;            // A-matrix row (pixel)
        const int koff = (lane >> 4) * 2;      // K sub-offset per half-wave
        const int nloc = lane & 15;            // D-matrix column (lane)
        const int hi   = lane >> 4;
#pragma unroll
        for (int nt = 0; nt < NT; ++nt) {
            v8f acc = {0.f, 0.f, 0.f, 0.f, 0.f, 0.f, 0.f, 0.f};
            const int ncol = nloc + nt * 16;
#pragma unroll
            for (int ck = 0; ck < CK; ++ck) {
                const int kk = ck * 4 + koff;
                v2f a2 = *(const v2f*)&Pw[mA * PS + kk];
                v2f b2;
                b2.x = sKTK[kk * Np + ncol];
                b2.y = sKTK[(kk + 1) * Np + ncol];
                acc = __builtin_amdgcn_wmma_f32_16x16x4_f32(
                    /*neg_a=*/false, a2, /*neg_b=*/false, b2,
                    /*c_mod=*/(short)0, acc, /*reuse_a=*/false, /*reuse_b=*/false);
            }
#pragma unroll
            for (int j = 0; j < 8; ++j) {
                // D VGPR j holds R[M = j + 8*hi][N = ncol]
                partial += acc[j] * Pw[(j + 8 * hi) * PS + ncol];
            }
        }
    }

    // ---- Phase 4: deterministic block reduction --------------------------
    sRed[tid] = partial;
    __syncthreads();
#pragma unroll
    for (int s = 128; s > 0; s >>= 1) {
        if (tid < s) sRed[tid] += sRed[tid + s];
        __syncthreads();
    }
    if (tid == 0) partials[blockIdx.x] = sRed[0] * inv_count;
}

// Final deterministic reduction of block partials into the scalar output.
__global__ __launch_bounds__(256) void reduce_partials_kernel(
    const float* __restrict__ partials, int n, float* __restrict__ out)
{
    __shared__ float s[256];
    float acc = 0.f;
    for (int i = threadIdx.x; i < n; i += 256) acc += partials[i];
    s[threadIdx.x] = acc;
    __syncthreads();
#pragma unroll
    for (int k = 128; k > 0; k >>= 1) {
        if (threadIdx.x < k) s[threadIdx.x] += s[threadIdx.x + k];
        __syncthreads();
    }
    if (threadIdx.x == 0) out[0] = s[0];
}

extern "C" void kernel_launch(void* const* d_in, const int* in_sizes, int n_in,
                              void* d_out, int out_size, void* d_ws, size_t ws_size,
                              hipStream_t stream)
{
    const float* flow0 = (const float*)d_in[0];  // (32, 2, 128, 128)
    const float* flow1 = (const float*)d_in[1];  // (32, 2, 256, 256)
    float* out = (float*)d_out;
    float* partials = (float*)d_ws;

    const int B = 32;

    // flow0: KZ=3, H=W=128  -> h=w=126, tilesX=8
    const int H0 = 128, W0 = 128;
    const int h0 = H0 - 3 + 1, w0 = W0 - 3 + 1;
    const int tiles0 = (w0 + 15) >> 4;
    const long waves0 = (long)B * 2 * h0 * tiles0;
    const int blocks0 = (int)((waves0 + 7) / 8);

    // flow1: KZ=5, H=W=256  -> h=w=252, tilesX=16
    const int H1 = 256, W1 = 256;
    const int h1 = H1 - 5 + 1, w1 = W1 - 5 + 1;
    const int tiles1 = (w1 + 15) >> 4;
    const long waves1 = (long)B * 2 * h1 * tiles1;
    const int blocks1 = (int)((waves1 + 7) / 8);

    affine_loss_kernel<3><<<blocks0, 256, 0, stream>>>(
        flow0, B, H0, W0, partials, 1.0f / (float)((long)B * h0 * w0));
    affine_loss_kernel<5><<<blocks1, 256, 0, stream>>>(
        flow1, B, H1, W1, partials + blocks0, 1.0f / (float)((long)B * h1 * w1));

    reduce_partials_kernel<<<1, 256, 0, stream>>>(partials, blocks0 + blocks1, out);
}